// Encoder_2422361554964
// MI455X (gfx1250) — compile-verified
//
#include <hip/hip_runtime.h>
#include <math.h>

typedef __attribute__((ext_vector_type(2))) float v2f;
typedef __attribute__((ext_vector_type(8))) float v8f;

#define Bn   64
#define Ln   2048
#define Dn   256
#define Gn   1024          // 4*D
#define NWG  32            // persistent workgroups
#define DW   8             // d-columns owned per WG
#define NCOL 32            // 4*DW gate columns per WG
#define KTOT 512           // concat K: x (256) + h (256)
#define BT_STRIDE 516      // 516 % 64 == 4 -> conflict-free b64 frag loads
#define GT_STRIDE 36

__global__ __launch_bounds__(256, 1)
void lstm_fused_kernel(const int* __restrict__ ctx, const float* __restrict__ emb,
                       const float* __restrict__ W,  const float* __restrict__ U,
                       const float* __restrict__ bia,
                       float* __restrict__ out,
                       float* __restrict__ hbuf0, float* __restrict__ hbuf1,
                       int* __restrict__ bar_cnt)
{
    __shared__ float Bt[NCOL][BT_STRIDE];   // [n][k]: k<256 -> W col, k>=256 -> U col
    __shared__ float gt[Bn][GT_STRIDE];     // gate tile 64 x 32
    __shared__ float cst[Bn][DW];           // cell state (WG-local columns)
    __shared__ float bs[NCOL];

    const int wg   = blockIdx.x;            // 0..31
    const int tid  = threadIdx.x;
    const int lane = tid & 31;
    const int wv   = tid >> 5;              // 0..7 (8 waves)
    const int lrow = lane & 15;
    const int lhi  = lane >> 4;

    // ---- one-time: stage transposed W/U slices + bias into LDS ----
    for (int idx = tid; idx < NCOL * KTOT; idx += 256) {
        int n = idx / KTOT;
        int k = idx - n * KTOT;
        int col = (n >> 3) * Dn + wg * DW + (n & 7);   // gate*256 + d
        Bt[n][k] = (k < Dn) ? W[k * Gn + col] : U[(k - Dn) * Gn + col];
    }
    if (tid < NCOL) {
        int col = (tid >> 3) * Dn + wg * DW + (tid & 7);
        bs[tid] = bia[col];
    }
    for (int idx = tid; idx < Bn * DW; idx += 256)
        cst[idx >> 3][idx & 7] = 0.0f;
    __syncthreads();

    const int mo = (wv & 3) * 16;           // C-tile M origin
    const int no = (wv >> 2) * 16;          // C-tile N origin
    const int m  = mo + lrow;               // batch row this lane reads for A
    const int nn = no + lrow;               // gate column this lane reads for B
    const float bv = bs[nn];

    // ---- prologue: acc = bias + x_0 @ W_slice (no dependence on h) ----
    v8f acc;
    {
        const int cv = ctx[m * Ln + 0];
        const float* xrow = emb + (long)cv * Dn;
        #pragma unroll
        for (int r = 0; r < 8; ++r) acc[r] = bv;
        #pragma unroll 4
        for (int kk = 0; kk < Dn; kk += 4) {
            int kA = kk + 2 * lhi;
            v2f a = *(const v2f*)(xrow + kA);
            v2f b = *(const v2f*)(&Bt[nn][kA]);
            acc = __builtin_amdgcn_wmma_f32_16x16x4_f32(false, a, false, b,
                                                        (short)0, acc, false, false);
        }
    }

    for (int t = 0; t < Ln; ++t) {
        const float* __restrict__ hr = (t & 1) ? hbuf1 : hbuf0;   // read h_{t-1}
        float*       __restrict__ hw = (t & 1) ? hbuf0 : hbuf1;   // publish h_t

        // ---- barrier WAIT phase: h_{t-1} from all WGs must be visible ----
        if (t > 0) {
            if (tid == 0) {
                const int target = NWG * t;
                while (__hip_atomic_load(bar_cnt, __ATOMIC_ACQUIRE,
                                         __HIP_MEMORY_SCOPE_AGENT) < target) {
                    __builtin_amdgcn_s_sleep(1);
                }
            }
            __syncthreads();
            __threadfence();
        }

        // ---- critical path: acc += h_{t-1} @ U_slice ----
        const float* hrow = hr + m * Dn;
        #pragma unroll 4
        for (int kk = 0; kk < Dn; kk += 4) {
            int kA = kk + 2 * lhi;
            v2f a = *(const v2f*)(hrow + kA);
            v2f b = *(const v2f*)(&Bt[nn][Dn + kA]);
            acc = __builtin_amdgcn_wmma_f32_16x16x4_f32(false, a, false, b,
                                                        (short)0, acc, false, false);
        }

        // scatter C-tile to LDS gate buffer (acc is dead afterwards)
        #pragma unroll
        for (int r = 0; r < 8; ++r)
            gt[mo + r + 8 * lhi][no + lrow] = acc[r];
        __syncthreads();

        // ---- elementwise LSTM cell for owned columns (512 (b,dl) pairs) ----
        for (int p = tid; p < Bn * DW; p += 256) {
            int b  = p >> 3;
            int dl = p & 7;
            float gi = gt[b][dl];
            float gf = gt[b][8  + dl];
            float gc = gt[b][16 + dl];
            float go = gt[b][24 + dl];
            float i = 1.0f / (1.0f + __expf(-gi));
            float f = 1.0f / (1.0f + __expf(-gf));
            float o = 1.0f / (1.0f + __expf(-go));
            float c_old = cst[b][dl];
            float c_new = f * c_old + i * tanhf(gc);
            float h_new = o * tanhf(c_new);

            int  d    = wg * DW + dl;
            bool msk  = (ctx[b * Ln + t] != 0);
            float h_old = hr[b * Dn + d];
            float h_out = msk ? h_new : h_old;
            if (msk) cst[b][dl] = c_new;
            hw[b * Dn + d] = h_out;                              // double-buffered publish
            __builtin_nontemporal_store(h_out, &out[((long)b * Ln + t) * Dn + d]);
        }
        __syncthreads();

        // ---- barrier ARRIVE phase: publish h_t, then overlap x GEMM ----
        __threadfence();
        if (tid == 0)
            __hip_atomic_fetch_add(bar_cnt, 1, __ATOMIC_RELEASE,
                                   __HIP_MEMORY_SCOPE_AGENT);

        // ---- overlapped with other WGs' sync: acc = bias + x_{t+1} @ W ----
        if (t + 1 < Ln) {
            const int cv = ctx[m * Ln + (t + 1)];
            const float* xrow = emb + (long)cv * Dn;
            #pragma unroll
            for (int r = 0; r < 8; ++r) acc[r] = bv;
            #pragma unroll 4
            for (int kk = 0; kk < Dn; kk += 4) {
                int kA = kk + 2 * lhi;
                v2f a = *(const v2f*)(xrow + kA);
                v2f b = *(const v2f*)(&Bt[nn][kA]);
                acc = __builtin_amdgcn_wmma_f32_16x16x4_f32(false, a, false, b,
                                                            (short)0, acc, false, false);
            }
        }
    }
}

extern "C" void kernel_launch(void* const* d_in, const int* in_sizes, int n_in,
                              void* d_out, int out_size, void* d_ws, size_t ws_size,
                              hipStream_t stream) {
    const int*   ctx = (const int*)  d_in[0];   // (B, L) int32
    const float* emb = (const float*)d_in[1];   // (VOCAB, D)
    const float* W   = (const float*)d_in[2];   // (D, 4D)
    const float* U   = (const float*)d_in[3];   // (D, 4D)
    const float* b   = (const float*)d_in[4];   // (4D)
    float* out = (float*)d_out;                 // (B, L, D)

    // workspace: h double buffer (2 x 64KB) + barrier counter
    float* hbuf0 = (float*)d_ws;
    float* hbuf1 = hbuf0 + Bn * Dn;
    int*   cnt   = (int*)((char*)d_ws + 2 * Bn * Dn * sizeof(float));

    hipMemsetAsync(d_ws, 0, 2 * Bn * Dn * sizeof(float) + 64, stream);
    lstm_fused_kernel<<<dim3(NWG), dim3(256), 0, stream>>>(
        ctx, emb, W, U, b, out, hbuf0, hbuf1, cnt);
}